// DGPool_8160437863162
// MI455X (gfx1250) — compile-verified
//
#include <hip/hip_runtime.h>

// ---------------------------------------------------------------------------
// DGPool for MI455X (gfx1250, wave32).
// ~400 MB total traffic -> ~17us roofline at 23.3 TB/s; compute trivial.
// Mat-vec runs on V_WMMA_F32_16X16X4_F32 with LDS-staged A tiles (async
// global->LDS double-buffered via global_load_async_to_lds_b128 +
// s_wait_asynccnt; register-pipelined synchronous fallback otherwise) and
// LDS-broadcast B. No atomics -> bitwise-deterministic across graph replays.
// ---------------------------------------------------------------------------

#define N_ROWS 8192
#define D_DIM  4096
#define K_SEL  4096          // k = N * 0.5
#define KM1    (K_SEL - 1)   // 4095

#define KSPLIT  4            // K-dimension split across blocks (occupancy)
#define KSLICE  (D_DIM / KSPLIT)   // 1024
#define KCHUNK  64           // K staged per pipeline stage
#define NCH     (KSLICE / KCHUNK)  // 16
#define TILE_STRIDE 68       // floats: 272B row stride = 16B multiple, no bank conflicts

typedef float v2f __attribute__((ext_vector_type(2)));
typedef float v8f __attribute__((ext_vector_type(8)));
typedef int   v4i __attribute__((ext_vector_type(4)));

#if __has_builtin(__builtin_amdgcn_global_load_async_to_lds_b128) && \
    __has_builtin(__builtin_amdgcn_s_wait_asynccnt)
#define USE_ASYNC 1
typedef __attribute__((address_space(1))) v4i* gv4i_p;  // global int4*
typedef __attribute__((address_space(3))) v4i* lv4i_p;  // LDS int4*
#else
#define USE_ASYNC 0
#endif

// ---------------------------------------------------------------------------
// ws layout (float offsets):
//   [3]              inv_norm
//   [64   .. 4160)   vn (normalized v)
//   [4352 .. 4384)   partS  (32 per-block score sums)
//   [4384 .. 4416)   partQ  (32 per-block score^2 sums)
//   [4416 .. 4448)   partPL (32 per-block pool-loss sums)
//   [8192 .. 40960)  rawp[4][8192] K-split partial scores
//   raw  aliases rawp split0 (written after all reads, same thread)
//   sig  aliases rawp split1 (written after dg_stats consumed rawp)
//   idx  aliases rawp split2
// ---------------------------------------------------------------------------

// K0: ||v|| and normalized v.
__global__ __launch_bounds__(256) void dg_vnorm(const float* __restrict__ v,
                                                float* __restrict__ wsf) {
  __shared__ float red[256];
  __shared__ float inv_s;
  float s = 0.f;
  for (int i = threadIdx.x; i < D_DIM; i += 256) {
    float t = v[i];
    s += t * t;
  }
  red[threadIdx.x] = s;
  __syncthreads();
  for (int off = 128; off > 0; off >>= 1) {
    if (threadIdx.x < off) red[threadIdx.x] += red[threadIdx.x + off];
    __syncthreads();
  }
  if (threadIdx.x == 0) {
    inv_s = 1.0f / (sqrtf(red[0]) + 1e-8f);
    wsf[3] = inv_s;
  }
  __syncthreads();
  const float iv = inv_s;
  for (int i = threadIdx.x; i < D_DIM; i += 256) wsf[64 + i] = v[i] * iv;
}

// ---------------------------------------------------------------------------
// K1: partial scores rawp[split][row] = x[row, k0:k0+1024] . vn[k0:k0+1024]
// via V_WMMA_F32_16X16X4_F32. Grid: 256 blocks = 64 row-groups x 4 K-splits.
// Each wave owns a 16-row strip; A tiles staged per-wave in LDS (no block
// barriers in the main loop), B broadcast from a block-shared vn slice.
// ---------------------------------------------------------------------------
__global__ __launch_bounds__(256) void dg_scores_wmma(
    const float* __restrict__ x, const float* __restrict__ vn,
    float* __restrict__ rawp) {
#if USE_ASYNC
  __shared__ float tiles[8][2][16 * TILE_STRIDE];  // double buffer per wave
#else
  __shared__ float tiles[8][1][16 * TILE_STRIDE];
#endif
  __shared__ float vnl[KSLICE];                    // 4KB vn slice for this split

  const int wave = threadIdx.x >> 5;
  const int lane = threadIdx.x & 31;
  const int row_group = blockIdx.x >> 2;           // 0..63
  const int split     = blockIdx.x & 3;            // 0..3
  const int k0        = split * KSLICE;
  const int row_base  = row_group * 128 + wave * 16;

  // block-shared copy of vn slice (one barrier total)
  ((float4*)vnl)[threadIdx.x] = ((const float4*)(vn + k0))[threadIdx.x];
  __syncthreads();

  const int m     = lane & 15;   // A row within strip
  const int khalf = lane >> 4;   // lanes 16-31 hold K+2 (16x4 f32 A layout)

  // staging lane map: call c covers rows 2c,2c+1; 16 lanes x 16B per row
  const int srow = lane >> 4;          // 0..1
  const int scol = (lane & 15) * 4;    // 0..60
  const float* gbase = x + (size_t)(row_base + srow) * D_DIM + k0 + scol;

  v8f acc = {0.f, 0.f, 0.f, 0.f, 0.f, 0.f, 0.f, 0.f};

  auto compute = [&](int ch, const float* t) {
    const int kc = ch * KCHUNK;
#pragma unroll
    for (int k4 = 0; k4 < KCHUNK; k4 += 4) {
      v2f a, b;
      a.x = t[m * TILE_STRIDE + k4 + 2 * khalf];
      a.y = t[m * TILE_STRIDE + k4 + 2 * khalf + 1];
      b.x = vnl[kc + k4 + 2 * khalf];      // broadcast across N columns
      b.y = vnl[kc + k4 + 2 * khalf + 1];
      acc = __builtin_amdgcn_wmma_f32_16x16x4_f32(
          false, a, false, b, (short)0, acc, false, false);
    }
  };

#if USE_ASYNC
  // ---- async global->LDS, double-buffered, per-wave s_wait_asynccnt ----
  auto issue = [&](int ch, float* tbuf) {
    const float* g = gbase + (size_t)ch * KCHUNK;
    float* l = tbuf + srow * TILE_STRIDE + scol;
#pragma unroll
    for (int c = 0; c < 8; ++c) {
      __builtin_amdgcn_global_load_async_to_lds_b128(
          (gv4i_p)(uintptr_t)(g + (size_t)2 * c * D_DIM),
          (lv4i_p)(uintptr_t)(l + 2 * c * TILE_STRIDE),
          0, 0);
    }
  };
  issue(0, tiles[wave][0]);
  for (int ch = 0; ch < NCH - 1; ++ch) {
    issue(ch + 1, tiles[wave][(ch + 1) & 1]);
    __builtin_amdgcn_s_wait_asynccnt(8);   // previous 8-op group complete
    asm volatile("" ::: "memory");
    compute(ch, tiles[wave][ch & 1]);
  }
  __builtin_amdgcn_s_wait_asynccnt(0);
  asm volatile("" ::: "memory");
  compute(NCH - 1, tiles[wave][(NCH - 1) & 1]);
#else
  // ---- synchronous fallback: register-pipelined loads, single LDS buffer.
  // Per-wave LDS is in-order: stores issued after this chunk's ds_loads
  // cannot bypass them, so one buffer is safe.
  float4 q[8];
  auto load_regs = [&](int ch) {
    const float* g = gbase + (size_t)ch * KCHUNK;
#pragma unroll
    for (int c = 0; c < 8; ++c) q[c] = *(const float4*)(g + (size_t)2 * c * D_DIM);
  };
  auto store_tile = [&]() {
    float* l = tiles[wave][0] + srow * TILE_STRIDE + scol;
#pragma unroll
    for (int c = 0; c < 8; ++c) *(float4*)(l + 2 * c * TILE_STRIDE) = q[c];
  };
  load_regs(0);
  store_tile();
  for (int ch = 0; ch < NCH; ++ch) {
    if (ch + 1 < NCH) load_regs(ch + 1);   // overlap with compute below
    compute(ch, tiles[wave][0]);
    if (ch + 1 < NCH) store_tile();
  }
#endif

  // D layout: VGPR r -> lanes0-15 (M=r,N=lane); lanes16-31 (M=8+r,N=lane-16)
  float* outp = rawp + (size_t)split * N_ROWS + row_base;
  if (lane == 0) {
#pragma unroll
    for (int r = 0; r < 8; ++r) outp[r] = acc[r];
  } else if (lane == 16) {
#pragma unroll
    for (int r = 0; r < 8; ++r) outp[8 + r] = acc[r];
  }
}

// ---------------------------------------------------------------------------
// K2: combine K-split partials (fixed order), per-block sum / sum-of-squares.
// ---------------------------------------------------------------------------
__global__ __launch_bounds__(256) void dg_stats(
    float* __restrict__ rawp, float* __restrict__ partS, float* __restrict__ partQ) {
  __shared__ float rs[256], rq[256];
  const int i = blockIdx.x * 256 + threadIdx.x;
  const float s = ((rawp[i] + rawp[N_ROWS + i]) +
                   (rawp[2 * N_ROWS + i] + rawp[3 * N_ROWS + i]));
  rawp[i] = s;  // combined raw score (aliases split0, written after all reads)
  rs[threadIdx.x] = s;
  rq[threadIdx.x] = s * s;
  __syncthreads();
  for (int off = 128; off > 0; off >>= 1) {
    if (threadIdx.x < off) {
      rs[threadIdx.x] += rs[threadIdx.x + off];
      rq[threadIdx.x] += rq[threadIdx.x + off];
    }
    __syncthreads();
  }
  if (threadIdx.x == 0) {
    partS[blockIdx.x] = rs[0];
    partQ[blockIdx.x] = rq[0];
  }
}

// ---------------------------------------------------------------------------
// K3: standardize + sigmoid + per-block pool-loss partial (deterministic).
// ---------------------------------------------------------------------------
__global__ __launch_bounds__(256) void dg_standardize(
    const float* __restrict__ raw, float* __restrict__ scores_out,
    float* __restrict__ sig, const float* __restrict__ partS,
    const float* __restrict__ partQ, float* __restrict__ partPL) {
  __shared__ float red[256];
  float sum = 0.f, sq = 0.f;
#pragma unroll
  for (int b = 0; b < 32; ++b) { sum += partS[b]; sq += partQ[b]; }
  const float mean = sum * (1.0f / N_ROWS);
  const float var  = fmaxf(sq * (1.0f / N_ROWS) - mean * mean, 0.0f);
  const float inv  = 1.0f / (sqrtf(var) + 1e-8f);

  const int i = blockIdx.x * 256 + threadIdx.x;
  const float s = (raw[i] - mean) * inv;
  scores_out[i] = s;
  const float sg = 1.0f / (1.0f + __expf(-s));
  sig[i] = sg;
  red[threadIdx.x] = sg * (1.0f - sg);
  __syncthreads();
  for (int off = 128; off > 0; off >>= 1) {
    if (threadIdx.x < off) red[threadIdx.x] += red[threadIdx.x + off];
    __syncthreads();
  }
  if (threadIdx.x == 0) partPL[blockIdx.x] = red[0];
}

// ---------------------------------------------------------------------------
// K4: bitonic sort of 8192 composite keys in 64 KB LDS.
// Key = (~monotone_u32(score) << 32) | index : ascending == score descending
// with stable index tie-break == exact jax top_k order.
// ---------------------------------------------------------------------------
__global__ __launch_bounds__(1024) void dg_sort(
    const float* __restrict__ scores, int* __restrict__ idx_out) {
  __shared__ unsigned long long keys[N_ROWS];
  for (int i = threadIdx.x; i < N_ROWS; i += 1024) {
    unsigned u = __float_as_uint(scores[i]);
    unsigned asc = (u & 0x80000000u) ? ~u : (u | 0x80000000u);
    keys[i] = ((unsigned long long)(~asc) << 32) | (unsigned)i;
  }
  __syncthreads();
  for (unsigned ksz = 2; ksz <= N_ROWS; ksz <<= 1) {
    for (unsigned j = ksz >> 1; j > 0; j >>= 1) {
      for (unsigned t = threadIdx.x; t < N_ROWS; t += 1024) {
        const unsigned ixj = t ^ j;
        if (ixj > t) {
          const bool up = ((t & ksz) == 0);
          const unsigned long long a = keys[t];
          const unsigned long long b = keys[ixj];
          if ((a > b) == up) { keys[t] = b; keys[ixj] = a; }
        }
      }
      __syncthreads();
    }
  }
  for (int i = threadIdx.x; i < K_SEL; i += 1024)
    idx_out[i] = (int)(keys[i] & 0xFFFFFFFFu);
}

// ---------------------------------------------------------------------------
// K5: new_x[r,:] = x[idx[r],:] * sig[idx[r]] -- float4 streaming + prefetch.
// ---------------------------------------------------------------------------
__global__ __launch_bounds__(256) void dg_gather(
    const float* __restrict__ x, const float* __restrict__ sig,
    const int* __restrict__ idx, float* __restrict__ newx) {
  const int row = blockIdx.x;
  const int src = idx[row];
  const float sc = sig[src];
  const float4* xr = (const float4*)(x + (size_t)src * D_DIM);
  float4* o = (float4*)(newx + (size_t)row * D_DIM);
  for (int c = threadIdx.x; c < D_DIM / 4; c += 256) {
    if (c + 256 < D_DIM / 4) __builtin_prefetch(xr + c + 256, 0, 0);
    float4 q = xr[c];
    o[c] = make_float4(q.x * sc, q.y * sc, q.z * sc, q.w * sc);
  }
}

// ---------------------------------------------------------------------------
// K6: edge index [2, k*(k-1)] + pool_loss finalize (fixed-order sum).
// ---------------------------------------------------------------------------
__global__ __launch_bounds__(256) void dg_edges(
    int* __restrict__ e, float* __restrict__ pl, const float* __restrict__ partPL) {
  const long long M = (long long)K_SEL * KM1;
  const int i = blockIdx.x;
  const long long base = (long long)i * KM1;
  for (int j = threadIdx.x; j < KM1; j += 256) {
    const int dst = j + (j >= i ? 1 : 0);
    e[base + j] = i;
    e[M + base + j] = dst;
  }
  if (blockIdx.x == 0 && threadIdx.x == 0) {
    float s = 0.f;
#pragma unroll
    for (int b = 0; b < 32; ++b) s += partPL[b];
    *pl = s * (1.0f / N_ROWS);
  }
}

// ---------------------------------------------------------------------------
extern "C" void kernel_launch(void* const* d_in, const int* in_sizes, int n_in,
                              void* d_out, int out_size, void* d_ws, size_t ws_size,
                              hipStream_t stream) {
  const float* x = (const float*)d_in[0];  // [8192, 4096]
  const float* v = (const float*)d_in[1];  // [4096, 1]

  float* out = (float*)d_out;
  float* newx   = out;                                       // 16,777,216 f32
  int*   edges  = (int*)(out + (size_t)16777216);            // 33,546,240 i32
  float* pl     = out + (size_t)16777216 + (size_t)33546240; // 1 f32
  float* scores = pl + 1;                                    // 8192 f32

  float* wsf    = (float*)d_ws;
  float* vn     = wsf + 64;
  float* partS  = wsf + 4352;
  float* partQ  = wsf + 4384;
  float* partPL = wsf + 4416;
  float* rawp   = wsf + 8192;          // 4 x 8192 partials
  float* raw    = rawp;                // combined (aliases split0)
  float* sig    = rawp + N_ROWS;       // aliases split1 (after dg_stats)
  int*   idx    = (int*)(rawp + 2 * N_ROWS);  // aliases split2

  dg_vnorm      <<<1,    256, 0, stream>>>(v, wsf);
  dg_scores_wmma<<<256,  256, 0, stream>>>(x, vn, rawp);
  dg_stats      <<<32,   256, 0, stream>>>(rawp, partS, partQ);
  dg_standardize<<<32,   256, 0, stream>>>(raw, scores, sig, partS, partQ, partPL);
  dg_sort       <<<1,   1024, 0, stream>>>(scores, idx);
  dg_gather     <<<K_SEL, 256, 0, stream>>>(x, sig, idx, newx);
  dg_edges      <<<K_SEL, 256, 0, stream>>>(edges, pl, partPL);
}